// Selective_Scan_Illum_48773648613387
// MI455X (gfx1250) — compile-verified
//
#include <hip/hip_runtime.h>

// Selective scan (illum-gated), B=64, L=8192, d_model=d_state=64.
// Chunked parallel scan: pass1 (per-chunk P,S) -> pass2 (combine) -> pass3
// (re-scan + output). Pass 3: x/illum streamed via GLOBAL_LOAD_ASYNC_TO_LDS
// (ASYNCcnt, double-buffered) and dot(h, A_log) via V_WMMA_F32_16X16X4_F32
// with A_log broadcast across the 16 B-columns so C[l][*] == dot[l].

constexpr int Bsz   = 64;
constexpr int Lseq  = 8192;
constexpr int Dm    = 64;           // d_model == d_state
constexpr int CHUNK = 128;
constexpr int NC    = Lseq / CHUNK; // 64 chunks per batch

typedef __attribute__((ext_vector_type(2))) float v2f;
typedef __attribute__((ext_vector_type(8))) float v8f;

// ---- CDNA5 async copy helpers (inline asm: portable across toolchains) -----
__device__ __forceinline__ unsigned lds_off(const void* p) {
    // generic -> LDS(AS3) addrspace cast; ptrtoint gives the 32-bit LDS byte
    // address expected by the async-load VDST operand.
    return (unsigned)(unsigned long long)(const __attribute__((address_space(3))) char*)p;
}
__device__ __forceinline__ void async_load_b128(unsigned lds_byte, const void* gaddr) {
    asm volatile("global_load_async_to_lds_b128 %0, %1, off"
                 :: "v"(lds_byte), "v"(gaddr) : "memory");
}
__device__ __forceinline__ void wait_async_le8() {
    asm volatile("s_wait_asynccnt 0x8" ::: "memory");
}
__device__ __forceinline__ void wait_async_0() {
    asm volatile("s_wait_asynccnt 0x0" ::: "memory");
}

// ---------------- Pass 1: per-chunk decay product P and local scan S ---------
__global__ __launch_bounds__(32) void k_pass1(
    const float* __restrict__ x, const float* __restrict__ il,
    const float* __restrict__ A_log, float* __restrict__ P, float* __restrict__ S)
{
    const int wg   = blockIdx.x;          // b * NC + c
    const int b    = wg >> 6;
    const int c    = wg & (NC - 1);
    const int lane = threadIdx.x;         // wave32
    const int s0   = lane * 2;

    const float nA0 = -__expf(A_log[s0]);
    const float nA1 = -__expf(A_log[s0 + 1]);

    const size_t base_l = (size_t)b * Lseq + (size_t)c * CHUNK;
    const float2* xr = reinterpret_cast<const float2*>(x + base_l * Dm) + lane;
    const float*  ib = il + base_l * 3;

    float h0 = 0.f, h1 = 0.f, p0 = 1.f, p1 = 1.f;
#pragma unroll 4
    for (int l = 0; l < CHUNK; ++l) {
        if ((l & 15) == 0)
            __builtin_prefetch(reinterpret_cast<const float*>(xr) + (l + 16) * Dm, 0, 1);
        const float pg = ib[3 * l + 0];
        const float pb = ib[3 * l + 1];
        const float de = ib[3 * l + 2];
        const float gamma = 0.3f + 0.7f / (1.f + __expf(-pg));
        const float beta  = -0.5f + pb;
        const float2 xv = xr[l * (Dm / 2)];
        const float d0 = __expf(nA0 * de);
        const float d1 = __expf(nA1 * de);
        h0 = h0 * d0 + (gamma * xv.x + beta);
        h1 = h1 * d1 + (gamma * xv.y + beta);
        p0 *= d0;  p1 *= d1;
    }
    reinterpret_cast<float2*>(P + (size_t)wg * Dm)[lane] = make_float2(p0, p1);
    reinterpret_cast<float2*>(S + (size_t)wg * Dm)[lane] = make_float2(h0, h1);
}

// ---------------- Pass 2: sequential combine over chunks --------------------
__global__ __launch_bounds__(64) void k_pass2(
    const float* __restrict__ P, const float* __restrict__ S, float* __restrict__ Hin)
{
    const int b = blockIdx.x;
    const int s = threadIdx.x;
    float h = 0.f;
    for (int c = 0; c < NC; ++c) {
        const size_t idx = ((size_t)b * NC + c) * Dm + s;
        Hin[idx] = h;                     // state at chunk entry
        h = h * P[idx] + S[idx];
    }
}

// ---------------- Pass 3: re-scan from Hin + emit y (async + WMMA) ----------
__global__ __launch_bounds__(32) void k_pass3(
    const float* __restrict__ x, const float* __restrict__ il,
    const float* __restrict__ A_log, const float* __restrict__ Dp,
    const float* __restrict__ Hin, float* __restrict__ y)
{
    __shared__ __align__(16) float xstage[2][16][64];   // async x double buffer
    __shared__ __align__(16) float ibuf[CHUNK * 3];     // async illum (whole chunk)
    __shared__ __align__(16) float hbuf[16][66];        // padded for A-reads
    __shared__ __align__(16) float xbuf[16][64];        // gated xg tile
    __shared__ __align__(16) float albuf[64];
    __shared__ float dotbuf[16];

    const int wg   = blockIdx.x;
    const int b    = wg >> 6;
    const int c    = wg & (NC - 1);
    const int lane = threadIdx.x;
    const int s0   = lane * 2;

    const float al0 = A_log[s0], al1 = A_log[s0 + 1];
    albuf[s0]     = al0;
    albuf[s0 + 1] = al1;
    const float nA0 = -__expf(al0), nA1 = -__expf(al1);

    const int  ocol = 4 * (lane & 15);                  // output columns per lane
    const int  orow = lane >> 4;
    const float4 d4 = *reinterpret_cast<const float4*>(Dp + ocol);

    const size_t base_l = (size_t)b * Lseq + (size_t)c * CHUNK;
    const char*  ibg    = reinterpret_cast<const char*>(il + base_l * 3);
    const char*  xg     = reinterpret_cast<const char*>(x + base_l * Dm);

    // --- stage whole chunk's illum (1536 B): 3 async b128 wave-instructions
    {
        const unsigned ldst = lds_off(&ibuf[0]) + 16u * lane;
#pragma unroll
        for (int j = 0; j < 3; ++j)
            async_load_b128(ldst + 512u * j, ibg + 16 * lane + 512 * j);
    }
    // --- stage x tile 0 (4 KB): 8 async b128 wave-instructions
#pragma unroll
    for (int j = 0; j < 8; ++j)
        async_load_b128(lds_off(&xstage[0][0][0]) + 16u * lane + 512u * j,
                        xg + 16 * lane + 512 * j);

    const float2 h2 = reinterpret_cast<const float2*>(Hin + (size_t)wg * Dm)[lane];
    float h0 = h2.x, h1 = h2.y;

    // WMMA f32 16x16x4 operand addressing (wave32):
    //  A (16x4): lanes 0-15 hold {K=0,K=1} of row M=lane, lanes 16-31 {K=2,K=3}
    //  B (4x16): lanes 0-15 hold rows {K=0,K=1} of col N=lane, lanes 16-31 {K=2,K=3}
    const int koff = (lane >> 4) * 2;
    const int arow = lane & 15;

    for (int t = 0; t < CHUNK / 16; ++t) {
        if (t < CHUNK / 16 - 1) {
            // issue next tile's async loads, then wait for current tile (+illum)
            const char*    gp   = xg + (size_t)(t + 1) * 16 * Dm * sizeof(float);
            const unsigned ldst = lds_off(&xstage[(t + 1) & 1][0][0]) + 16u * lane;
#pragma unroll
            for (int j = 0; j < 8; ++j)
                async_load_b128(ldst + 512u * j, gp + 16 * lane + 512 * j);
            wait_async_le8();   // in-order completion => current tile resident
        } else {
            wait_async_0();
        }

        const float (*xs)[64] = xstage[t & 1];
#pragma unroll
        for (int l = 0; l < 16; ++l) {
            const int ll = t * 16 + l;
            const float pg = ibuf[3 * ll + 0];
            const float pb = ibuf[3 * ll + 1];
            const float de = ibuf[3 * ll + 2];
            const float gamma = 0.3f + 0.7f / (1.f + __expf(-pg));
            const float beta  = -0.5f + pb;
            const float2 xv = *reinterpret_cast<const float2*>(&xs[l][s0]);
            const float d0 = __expf(nA0 * de);
            const float d1 = __expf(nA1 * de);
            const float xg0 = gamma * xv.x + beta;
            const float xg1 = gamma * xv.y + beta;
            h0 = h0 * d0 + xg0;
            h1 = h1 * d1 + xg1;
            *reinterpret_cast<float2*>(&hbuf[l][s0]) = make_float2(h0, h1);
            *reinterpret_cast<float2*>(&xbuf[l][s0]) = make_float2(xg0, xg1);
        }

        // dot[l] = sum_k h[l][k] * A_log[k]  via 16 accumulating 16x16x4 WMMAs.
        // Every B column is A_log[k], so every C column equals dot[l].
        v8f acc = {0.f, 0.f, 0.f, 0.f, 0.f, 0.f, 0.f, 0.f};
#pragma unroll
        for (int kk = 0; kk < 16; ++kk) {
            const int kb = kk * 4 + koff;
            const float2 av = *reinterpret_cast<const float2*>(&hbuf[arow][kb]);
            const float2 bv = *reinterpret_cast<const float2*>(&albuf[kb]);
            v2f a  = {av.x, av.y};
            v2f bb = {bv.x, bv.y};
            acc = __builtin_amdgcn_wmma_f32_16x16x4_f32(
                /*neg_a=*/false, a, /*neg_b=*/false, bb,
                /*c_mod=*/(short)0, acc, /*reuse_a=*/false, /*reuse_b=*/false);
        }

        // C layout: VGPR j holds M=j (lanes 0-15) and M=j+8 (lanes 16-31).
        // One representative lane per half broadcasts dot[l] through LDS.
        if (lane == 0) {
#pragma unroll
            for (int j = 0; j < 8; ++j) dotbuf[j] = acc[j];
        }
        if (lane == 16) {
#pragma unroll
            for (int j = 0; j < 8; ++j) dotbuf[8 + j] = acc[j];
        }

        // y[row][d] = dot[row] + D[d]*xg[row][d]; lane covers 4 contiguous d's,
        // two rows per b128 store wave-instruction.
#pragma unroll
        for (int l2 = 0; l2 < 16; l2 += 2) {
            const int row = l2 + orow;
            const float  dt = dotbuf[row];
            const float4 xv = *reinterpret_cast<const float4*>(&xbuf[row][ocol]);
            float* yp = y + (base_l + (size_t)t * 16 + row) * Dm + ocol;
            *reinterpret_cast<float4*>(yp) =
                make_float4(dt + d4.x * xv.x, dt + d4.y * xv.y,
                            dt + d4.z * xv.z, dt + d4.w * xv.w);
        }
    }
}

extern "C" void kernel_launch(void* const* d_in, const int* in_sizes, int n_in,
                              void* d_out, int out_size, void* d_ws, size_t ws_size,
                              hipStream_t stream) {
    const float* x  = (const float*)d_in[0];   // [64, 8192, 64]
    const float* il = (const float*)d_in[1];   // [64, 8192, 3]
    const float* al = (const float*)d_in[2];   // [64]
    const float* Dp = (const float*)d_in[3];   // [64]
    float* y = (float*)d_out;                  // [64, 8192, 64]

    const size_t per = (size_t)Bsz * NC * Dm;  // 262144 floats = 1 MB each
    float* P = (float*)d_ws;
    float* S = P + per;
    float* H = S + per;

    k_pass1<<<Bsz * NC, 32, 0, stream>>>(x, il, al, P, S);
    k_pass2<<<Bsz, 64, 0, stream>>>(P, S, H);
    k_pass3<<<Bsz * NC, 32, 0, stream>>>(x, il, al, Dp, H, y);
}